// GraphEncoder_7876970020898
// MI455X (gfx1250) — compile-verified
//
#include <hip/hip_runtime.h>

typedef __attribute__((ext_vector_type(2))) float v2f;
typedef __attribute__((ext_vector_type(8))) float v8f;

// Problem constants (fixed by the reference)
constexpr int NN   = 50000;   // nodes
constexpr int EE   = 800000;  // edges (before self loops)
constexpr int INC  = 128;
constexpr int HID  = 96;
constexpr int OUTC = 48;

// ---------------------------------------------------------------------------
// init: deg = 1.0 (self loop), zero layer-1 accumulator, zero final output
// ---------------------------------------------------------------------------
__global__ void k_init(float* __restrict__ deg, float* __restrict__ agg1,
                       float* __restrict__ out) {
  size_t i = (size_t)blockIdx.x * blockDim.x + threadIdx.x;
  size_t stride = (size_t)gridDim.x * blockDim.x;
  for (size_t t = i; t < (size_t)NN * HID; t += stride) {
    agg1[t] = 0.0f;
    if (t < (size_t)NN) deg[t] = 1.0f;
    if (t < (size_t)NN * OUTC) out[t] = 0.0f;
  }
}

// ---------------------------------------------------------------------------
// degree: deg[dst[e]] += 1  (f32 atomics, L2-resident)
// ---------------------------------------------------------------------------
__global__ void k_degree(const long long* __restrict__ dst, float* __restrict__ deg) {
  int i = blockIdx.x * blockDim.x + threadIdx.x;
  if (i < EE) atomicAdd(deg + (int)dst[i], 1.0f);
}

// deg -> deg^{-1/2} in place (deg >= 1 always, self loop included)
__global__ void k_rsqrt(float* __restrict__ deg) {
  int i = blockIdx.x * blockDim.x + threadIdx.x;
  if (i < NN) deg[i] = rsqrtf(deg[i]);
}

// ---------------------------------------------------------------------------
// fp32 WMMA GEMM: C[M x N] = A[M x K] * B[K x N]
// One wave per 16-row stripe; NT = N/16 column tiles per wave so the A
// fragment is loaded once per k-step and reused (v_wmma_f32_16x16x4_f32).
//
// A-frag layout (32-bit A 16x4): lane l<16 -> row=l, {K=k0,k0+1};
//                                lane l>=16 -> row=l-16, {K=k0+2,k0+3}
// B-frag mirrors it; C/D: VGPR i -> M = i + 8*(lane>=16), N = lane&15.
// ---------------------------------------------------------------------------
template <int K, int N, int NT>
__global__ void k_gemm_wmma(const float* __restrict__ A, const float* __restrict__ B,
                            float* __restrict__ C) {
  const int lane = threadIdx.x;        // 0..31, block = exactly one wave
  const int half = lane >> 4;          // 0 or 1
  const int l    = lane & 15;
  const int row  = blockIdx.x * 16 + l;

  v8f acc[NT];
#pragma unroll
  for (int j = 0; j < NT; ++j) acc[j] = (v8f){0, 0, 0, 0, 0, 0, 0, 0};

#pragma unroll 4
  for (int k0 = 0; k0 < K; k0 += 4) {
    const int kk = k0 + half * 2;
    v2f a = *(const v2f*)(A + (size_t)row * K + kk);   // contiguous 8B load
#pragma unroll
    for (int j = 0; j < NT; ++j) {
      v2f b;
      b.x = B[(size_t)(kk + 0) * N + j * 16 + l];
      b.y = B[(size_t)(kk + 1) * N + j * 16 + l];
      acc[j] = __builtin_amdgcn_wmma_f32_16x16x4_f32(
          /*neg_a=*/false, a, /*neg_b=*/false, b,
          /*c_mod=*/(short)0, acc[j], /*reuse_a=*/false, /*reuse_b=*/false);
    }
  }

#pragma unroll
  for (int j = 0; j < NT; ++j) {
#pragma unroll
    for (int i = 0; i < 8; ++i) {
      const int r = blockIdx.x * 16 + i + half * 8;
      C[(size_t)r * N + j * 16 + l] = acc[j][i];
    }
  }
}

// ---------------------------------------------------------------------------
// scatter: for each (virtual) edge e in [0, E+N):
//   e <  E : s=src[e], d=dst[e]
//   e >= E : self loop s=d=e-E
//   out[d][:] += H[s][:] * dinv[s]*dinv[d]     (f32 atomics, L2-resident)
// One wave per edge; F/32 features per lane.
// ---------------------------------------------------------------------------
template <int F>
__global__ void k_scatter(const float* __restrict__ H, const long long* __restrict__ src,
                          const long long* __restrict__ dst, const float* __restrict__ dinv,
                          float* __restrict__ out) {
  const int lane = threadIdx.x & 31;
  const long long e =
      (long long)blockIdx.x * (blockDim.x >> 5) + (threadIdx.x >> 5);
  if (e >= (long long)EE + NN) return;
  int s, d;
  if (e < EE) {
    s = (int)src[e];
    d = (int)dst[e];
  } else {
    s = d = (int)(e - EE);
  }
  const float w = dinv[s] * dinv[d];
  const float* hs = H + (size_t)s * F;
  float* od = out + (size_t)d * F;
#pragma unroll
  for (int f = lane; f < F; f += 32) {
    atomicAdd(od + f, hs[f] * w);
  }
}

// h1 = relu(agg1 + b1)
__global__ void k_bias_relu(const float* __restrict__ agg, const float* __restrict__ bias,
                            float* __restrict__ H) {
  size_t i = (size_t)blockIdx.x * blockDim.x + threadIdx.x;
  if (i < (size_t)NN * HID) {
    const int f = (int)(i % HID);
    const float v = agg[i] + bias[f];
    H[i] = v > 0.0f ? v : 0.0f;
  }
}

// out += b2
__global__ void k_bias(float* __restrict__ out, const float* __restrict__ bias) {
  size_t i = (size_t)blockIdx.x * blockDim.x + threadIdx.x;
  if (i < (size_t)NN * OUTC) out[i] += bias[(int)(i % OUTC)];
}

// ---------------------------------------------------------------------------
extern "C" void kernel_launch(void* const* d_in, const int* in_sizes, int n_in,
                              void* d_out, int out_size, void* d_ws, size_t ws_size,
                              hipStream_t stream) {
  const float*     x   = (const float*)d_in[0];
  const long long* ei  = (const long long*)d_in[1];   // [2, E] int64
  const float*     W1  = (const float*)d_in[2];
  const float*     b1  = (const float*)d_in[3];
  const float*     W2  = (const float*)d_in[4];
  const float*     b2  = (const float*)d_in[5];
  float*           out = (float*)d_out;

  const long long* src = ei;
  const long long* dst = ei + EE;

  // Workspace layout (floats): deg | H1 | AGG1 | G2   (~48.3 MB total)
  float* deg  = (float*)d_ws;
  float* H1   = deg + 65536;                 // 256 KB pad for deg/dinv
  float* AGG1 = H1 + (size_t)NN * HID;       // 19.2 MB
  float* G2   = AGG1 + (size_t)NN * HID;     // 19.2 MB ; G2 = 9.6 MB

  // 1) init deg=1, zero AGG1, zero out
  k_init<<<1024, 256, 0, stream>>>(deg, AGG1, out);

  // 2) degree accumulation over dst
  k_degree<<<(EE + 255) / 256, 256, 0, stream>>>(dst, deg);

  // 3) deg -> rsqrt(deg)
  k_rsqrt<<<(NN + 255) / 256, 256, 0, stream>>>(deg);

  // 4) H1 = X @ W1   (fp32 WMMA, 16-row stripes x 6 col tiles)
  k_gemm_wmma<INC, HID, HID / 16><<<NN / 16, 32, 0, stream>>>(x, W1, H1);

  // 5) AGG1 = scatter(H1[src]*norm -> dst), incl. self loops
  const long long totE = (long long)EE + NN;
  k_scatter<HID><<<(unsigned)((totE + 7) / 8), 256, 0, stream>>>(H1, src, dst, deg, AGG1);

  // 6) H1 = relu(AGG1 + b1)   (reuse H1 buffer)
  k_bias_relu<<<(NN * HID + 255) / 256, 256, 0, stream>>>(AGG1, b1, H1);

  // 7) G2 = H1 @ W2   (fp32 WMMA, 3 col tiles)
  k_gemm_wmma<HID, OUTC, OUTC / 16><<<NN / 16, 32, 0, stream>>>(H1, W2, G2);

  // 8) out = scatter(G2[src]*norm -> dst)
  k_scatter<OUTC><<<(unsigned)((totE + 7) / 8), 256, 0, stream>>>(G2, src, dst, deg, out);

  // 9) out += b2
  k_bias<<<(NN * OUTC + 255) / 256, 256, 0, stream>>>(out, b2);
}